// Encoder_35304631174306
// MI455X (gfx1250) — compile-verified
//
#include <hip/hip_runtime.h>
#include <hip/hip_bf16.h>
#include <math.h>

typedef __attribute__((ext_vector_type(2))) float v2f;
typedef __attribute__((ext_vector_type(8))) float v8f;

#define B_SZ 32
#define S_SZ 512
#define E_SZ 512
#define U_SZ 1024
#define G_SZ 4096          /* 4*U_SZ */
#define NROW 16384         /* B*S    */
#define NWG_SCAN 32

// ---- CDNA5 async global->LDS copies (ASYNCcnt-tracked DMA) --------------
__device__ __forceinline__ void async_g2l_b128(unsigned lds_off, const void* g) {
    asm volatile("global_load_async_to_lds_b128 %0, %1, off"
                 :: "v"(lds_off), "v"((unsigned long long)(uintptr_t)g)
                 : "memory");
}
__device__ __forceinline__ void async_g2l_b64(unsigned lds_off, const void* g) {
    asm volatile("global_load_async_to_lds_b64 %0, %1, off"
                 :: "v"(lds_off), "v"((unsigned long long)(uintptr_t)g)
                 : "memory");
}
#define WAIT_ASYNC(n) asm volatile("s_wait_asynccnt " #n ::: "memory")

__device__ __forceinline__ unsigned lds_off_of(const void* p) {
    return (unsigned)(uintptr_t)p;       // LDS aperture: addr[31:0] = LDS offset
}

// ---------------------------------------------------------------- init
__global__ void init_kernel(float* hbuf, unsigned* cnt) {
    int i = blockIdx.x * blockDim.x + threadIdx.x;
    if (i < 2 * B_SZ * U_SZ) hbuf[i] = 0.0f;
    if (i == 0) *cnt = 0u;
}

// ------------------------------------------------- gather + GEMM (xW)
// xW[r][g] = sum_e emb[tok[r]][e] * W[e][g] + bias[g]
// Block tile 64x64, 8 waves, each wave: 16x32 strip (2 WMMA C tiles).
// Double-buffered LDS tiles filled by async global->LDS DMA.
__global__ __launch_bounds__(256)
void xw_gemm_kernel(const int* __restrict__ tokens,
                    const float* __restrict__ emb,
                    const float* __restrict__ W,
                    const float* __restrict__ bias,
                    float* __restrict__ xw) {
    __shared__ float As[2][64 * 20];   // 64 rows x 16 k, stride 20
    __shared__ float Bs[2][16 * 68];   // 16 k  x 64 n, stride 68

    const int tid  = threadIdx.x;
    const int lane = tid & 31;
    const int wave = tid >> 5;
    const int r0   = (blockIdx.x >> 6) * 64;   // 256 row tiles
    const int n0   = (blockIdx.x & 63) * 64;   // 64 col tiles

    const int rt = wave & 3;              // row tile 0..3
    const int ch = wave >> 2;             // col half 0..1
    const int m  = lane & 15;             // M (and N) index within tile
    const int kk = (lane >> 4) << 1;      // K sub-offset: 0 or 2

    v8f c0 = {}; v8f c1 = {};

    // A staging: thread -> (row, 4 consecutive k); B: (k row, 4 consecutive n)
    const int lr = tid >> 2;              // 0..63
    const int lk = (tid & 3) << 2;        // 0,4,8,12
    const int tok = tokens[r0 + lr];
    const float* erow = emb + (long)tok * E_SZ;
    const int bk = tid >> 4;              // 0..15
    const int bn = (tid & 15) << 2;       // 0..60

    const unsigned aoff[2] = { lds_off_of(&As[0][lr * 20 + lk]),
                               lds_off_of(&As[1][lr * 20 + lk]) };
    const unsigned boff[2] = { lds_off_of(&Bs[0][bk * 68 + bn]),
                               lds_off_of(&Bs[1][bk * 68 + bn]) };

    // prologue: chunk 0 into buffer 0
    async_g2l_b128(aoff[0], erow + 0 + lk);
    async_g2l_b128(boff[0], W + (long)(0 + bk) * G_SZ + n0 + bn);

    for (int it = 0; it < E_SZ / 16; ++it) {
        const int cur = it & 1;
        const int kb  = it * 16;
        if (kb + 16 < E_SZ) {      // prefetch next chunk into other buffer
            async_g2l_b128(aoff[cur ^ 1], erow + (kb + 16) + lk);
            async_g2l_b128(boff[cur ^ 1], W + (long)(kb + 16 + bk) * G_SZ + n0 + bn);
            WAIT_ASYNC(0x2);       // current chunk's 2 copies complete
        } else {
            WAIT_ASYNC(0x0);
        }
        __syncthreads();
#pragma unroll
        for (int i = 0; i < 4; ++i) {
            const int kp = 4 * i + kk;
            v2f a = *(const v2f*)(&As[cur][(rt * 16 + m) * 20 + kp]);
            v2f b0, b1;
            b0.x = Bs[cur][(kp    ) * 68 + ch * 32 + m];
            b0.y = Bs[cur][(kp + 1) * 68 + ch * 32 + m];
            b1.x = Bs[cur][(kp    ) * 68 + ch * 32 + 16 + m];
            b1.y = Bs[cur][(kp + 1) * 68 + ch * 32 + 16 + m];
            c0 = __builtin_amdgcn_wmma_f32_16x16x4_f32(false, a, false, b0,
                                                       (short)0, c0, false, false);
            c1 = __builtin_amdgcn_wmma_f32_16x16x4_f32(false, a, false, b1,
                                                       (short)0, c1, false, false);
        }
        __syncthreads();           // buffer free before it is refilled
    }

    // Epilogue: C layout => row = r + 8*(lane>=16), col = lane%16
    const int row_base = r0 + rt * 16 + ((lane >> 4) << 3);
    const int col0 = n0 + ch * 32 + m;
    const int col1 = col0 + 16;
    const float bb0 = bias[col0];
    const float bb1 = bias[col1];
#pragma unroll
    for (int r = 0; r < 8; ++r) {
        xw[(long)(row_base + r) * G_SZ + col0] = c0[r] + bb0;
        xw[(long)(row_base + r) * G_SZ + col1] = c1[r] + bb1;
    }
}

// ------------------------------------------------------- grid barrier
__device__ inline void grid_barrier(unsigned* cnt, unsigned target) {
    __threadfence();          // make h writes device-visible
    __syncthreads();
    if (threadIdx.x == 0) {
        __hip_atomic_fetch_add(cnt, 1u, __ATOMIC_RELEASE, __HIP_MEMORY_SCOPE_AGENT);
        while (__hip_atomic_load(cnt, __ATOMIC_ACQUIRE, __HIP_MEMORY_SCOPE_AGENT) < target) {
            __builtin_amdgcn_s_sleep(2);
        }
    }
    __syncthreads();
    __threadfence();          // acquire side for everyone
}

// ------------------------------------------------------ persistent scan
// 32 WGs; WG owns hidden cols [u0, u0+32) => gate cols {g*1024+u0..+31}.
// Per step: z(32x128) = h(32x1024) @ U cols, gates, c in LDS, h ping-pong.
// U/H tiles double-buffered via async global->LDS DMA.
__global__ __launch_bounds__(256)
void lstm_scan_kernel(const int* __restrict__ tokens,
                      const float* __restrict__ U,
                      const float* __restrict__ xw,
                      float* __restrict__ hbuf,      // 2 x 32 x 1024
                      unsigned* __restrict__ cnt,
                      float* __restrict__ out) {
    __shared__ float Hs[2][32 * 20];    // 32 rows x 16 k
    __shared__ float Us[2][16 * 132];   // 16 k x 128 cols
    __shared__ float Zs[32 * 132];      // z tile 32 x 128
    __shared__ float Cs[32 * 32];       // cell state chunk

    const int tid  = threadIdx.x;
    const int lane = tid & 31;
    const int wave = tid >> 5;            // col tile ct = wave (0..7)
    const int u0   = blockIdx.x << 5;

    const int m  = lane & 15;
    const int kk = (lane >> 4) << 1;

    // H staging: thread -> (row, 2 k); U staging: (k row, 8 cols)
    const int hrow = tid >> 3;            // 0..31
    const int hk   = (tid & 7) << 1;      // 0..14
    const int ur = tid >> 4;              // 0..15
    const int uc = (tid & 15) << 3;       // 0..120 (stays within one gate)
    const long ugcol = (long)(uc >> 5) * U_SZ + u0 + (uc & 31);

    const unsigned hoff[2] = { lds_off_of(&Hs[0][hrow * 20 + hk]),
                               lds_off_of(&Hs[1][hrow * 20 + hk]) };
    const unsigned uoff0[2] = { lds_off_of(&Us[0][ur * 132 + uc]),
                                lds_off_of(&Us[1][ur * 132 + uc]) };
    const unsigned uoff1[2] = { lds_off_of(&Us[0][ur * 132 + uc + 4]),
                                lds_off_of(&Us[1][ur * 132 + uc + 4]) };

    for (int i = tid; i < 32 * 32; i += 256) Cs[i] = 0.0f;
    __syncthreads();

    for (int t = 0; t < S_SZ; ++t) {
        const float* __restrict__ h_r = hbuf + (t & 1) * (B_SZ * U_SZ);
        float* __restrict__       h_w = hbuf + ((t & 1) ^ 1) * (B_SZ * U_SZ);

        // prologue: chunk 0 into buffer 0 (3 async ops/wave)
        {
            const float* up = U + (long)ur * G_SZ + ugcol;
            async_g2l_b64 (hoff[0],  h_r + hrow * U_SZ + hk);
            async_g2l_b128(uoff0[0], up);
            async_g2l_b128(uoff1[0], up + 4);
        }

        v8f c0 = {}; v8f c1 = {};
        for (int it = 0; it < U_SZ / 16; ++it) {
            const int cur = it & 1;
            const int kb  = it * 16;
            if (kb + 16 < U_SZ) {
                const float* up = U + (long)(kb + 16 + ur) * G_SZ + ugcol;
                async_g2l_b64 (hoff[cur ^ 1],  h_r + hrow * U_SZ + kb + 16 + hk);
                async_g2l_b128(uoff0[cur ^ 1], up);
                async_g2l_b128(uoff1[cur ^ 1], up + 4);
                WAIT_ASYNC(0x3);     // current chunk's 3 copies complete
            } else {
                WAIT_ASYNC(0x0);
            }
            __syncthreads();
#pragma unroll
            for (int i = 0; i < 4; ++i) {
                const int kp = 4 * i + kk;
                v2f a0 = *(const v2f*)(&Hs[cur][m * 20 + kp]);          // rows 0..15
                v2f a1 = *(const v2f*)(&Hs[cur][(16 + m) * 20 + kp]);   // rows 16..31
                v2f b;
                b.x = Us[cur][(kp    ) * 132 + wave * 16 + m];
                b.y = Us[cur][(kp + 1) * 132 + wave * 16 + m];
                c0 = __builtin_amdgcn_wmma_f32_16x16x4_f32(false, a0, false, b,
                                                           (short)0, c0, false, false);
                c1 = __builtin_amdgcn_wmma_f32_16x16x4_f32(false, a1, false, b,
                                                           (short)0, c1, false, false);
            }
            __syncthreads();         // buffer free before refill
        }

        // dump z tiles to LDS
        {
            const int zr = (lane >> 4) << 3;
#pragma unroll
            for (int r = 0; r < 8; ++r) {
                Zs[(zr + r) * 132 + wave * 16 + m]      = c0[r];
                Zs[(16 + zr + r) * 132 + wave * 16 + m] = c1[r];
            }
        }
        __syncthreads();

        // gate math: thread handles 4 (b, u_local) cells
#pragma unroll
        for (int j = 0; j < 4; ++j) {
            const int idx = tid + j * 256;
            const int bb  = idx >> 5;
            const int ul  = idx & 31;
            const long xb = ((long)bb * S_SZ + t) * G_SZ + u0 + ul;
            const float zi = Zs[bb * 132 + ul]       + xw[xb];
            const float zf = Zs[bb * 132 + 32 + ul]  + xw[xb + 1024];
            const float zg = Zs[bb * 132 + 64 + ul]  + xw[xb + 2048];
            const float zo = Zs[bb * 132 + 96 + ul]  + xw[xb + 3072];
            const float iv = tanhf(zi), fv = tanhf(zf);
            const float gv = tanhf(zg), ov = tanhf(zo);
            const float c_old = Cs[bb * 32 + ul];
            const float h_old = h_r[bb * U_SZ + u0 + ul];
            const float c_new = fv * c_old + iv * gv;
            const float h_new = ov * tanhf(c_new);
            const bool upd = (tokens[bb * S_SZ + t] != 0);
            const float cv = upd ? c_new : c_old;
            const float hv = upd ? h_new : h_old;
            Cs[bb * 32 + ul] = cv;
            h_w[bb * U_SZ + u0 + ul] = hv;
        }
        grid_barrier(cnt, (unsigned)(NWG_SCAN * (t + 1)));
    }

    // final h is in hbuf[0] (512 steps, even count). out = [h, h, c]
    const float* hf = hbuf;
#pragma unroll
    for (int j = 0; j < 4; ++j) {
        const int idx = tid + j * 256;
        const int bb  = idx >> 5;
        const int ul  = idx & 31;
        const float hv = hf[bb * U_SZ + u0 + ul];
        const float cv = Cs[bb * 32 + ul];
        out[(long)bb * U_SZ + u0 + ul]              = hv;
        out[32768 + (long)bb * U_SZ + u0 + ul]      = hv;
        out[65536 + (long)bb * U_SZ + u0 + ul]      = cv;
    }
}

// ---------------------------------------------------------------- launch
extern "C" void kernel_launch(void* const* d_in, const int* in_sizes, int n_in,
                              void* d_out, int out_size, void* d_ws, size_t ws_size,
                              hipStream_t stream) {
    const int*   tokens = (const int*)d_in[0];
    const float* emb    = (const float*)d_in[1];
    const float* W      = (const float*)d_in[2];
    const float* U      = (const float*)d_in[3];
    const float* bias   = (const float*)d_in[4];
    float* out = (float*)d_out;

    float*    xw   = (float*)d_ws;                       // 16384*4096 f32 (256 MiB)
    float*    hbuf = xw + (size_t)NROW * G_SZ;           // 2*32*1024 f32
    unsigned* cnt  = (unsigned*)(hbuf + 2 * B_SZ * U_SZ);

    init_kernel<<<(2 * B_SZ * U_SZ + 255) / 256, 256, 0, stream>>>(hbuf, cnt);
    xw_gemm_kernel<<<256 * 64, 256, 0, stream>>>(tokens, emb, W, bias, xw);
    lstm_scan_kernel<<<NWG_SCAN, 256, 0, stream>>>(tokens, U, xw, hbuf, cnt, out);
}